// DGCFModel_40913858461856
// MI455X (gfx1250) — compile-verified
//
#include <hip/hip_runtime.h>
#include <hip/hip_bf16.h>
#include <stdint.h>

typedef __attribute__((ext_vector_type(2))) float v2f;
typedef __attribute__((ext_vector_type(8))) float v8f;
// payload type for async b64 builtin (from hipcc diagnostic: v2i pointers)
typedef int async_b64_t __attribute__((__vector_size__(2 * sizeof(int))));

#if defined(__HIP_DEVICE_COMPILE__) && __has_builtin(__builtin_amdgcn_global_load_async_to_lds_b64)
#define USE_ASYNC_LDS 1
#else
#define USE_ASYNC_LDS 0
#endif

// ---------------------------------------------------------------------------
// DGCF on MI455X. Node feature arrays (38.4MB) are L2-resident; kernels are
// gather/atomic-scatter bound. WMMA f32 16x16x4 performs the segmented
// per-intent dot-product reductions for the routing attention; edge indices
// are staged wave-locally through LDS with async loads.
// ---------------------------------------------------------------------------

__device__ __forceinline__ const float* x0_row(const float* Gu, const float* Gi,
                                               int nu, int n) {
    return (n < nu) ? (Gu + (size_t)n * 64) : (Gi + (size_t)(n - nu) * 64);
}

// Branch-free tanh: hardware v_tanh_f32 if available, else exp-based identity.
// Keeps EXEC untouched around the WMMA instructions (ISA: EXEC must be all 1s).
__device__ __forceinline__ float fast_tanh(float x) {
#if defined(__HIP_DEVICE_COMPILE__) && __has_builtin(__builtin_amdgcn_tanhf)
    return __builtin_amdgcn_tanhf(x);
#else
    float t = __expf(2.0f * x);
    return (t - 1.0f) * __builtin_amdgcn_rcpf(t + 1.0f);
#endif
}

// ---- degree count (iteration 1: s == 0.25 uniformly, deg1 = 0.25 * count) ----
__global__ void k_degcnt(const int64_t* __restrict__ uIdx,
                         const int64_t* __restrict__ iIdx,
                         float* __restrict__ deg1, int nInter) {
    int j = blockIdx.x * blockDim.x + threadIdx.x;
    if (j >= nInter) return;
    int u = (int)uIdx[j];
    int i = (int)iIdx[j];
    atomicAdd(&deg1[u], 1.0f);
    atomicAdd(&deg1[i], 1.0f);
}

// ---- in-place d^{-1/2}: a[t] = a[t]*scale > 0 ? rsqrt(a[t]*scale) : 0 ----
__global__ void k_rsqrt(float* __restrict__ a, int n, float scale) {
    int t = blockIdx.x * blockDim.x + threadIdx.x;
    if (t >= n) return;
    float c = a[t] * scale;
    a[t] = (c > 0.0f) ? rsqrtf(c) : 0.0f;
}

// ---- propagation, iteration 1 (intent-independent, symmetric norm) ----
// one wave per interaction; each lane handles 2 of the 64 dims, both directions
__global__ void __launch_bounds__(256)
k_prop1(const int64_t* __restrict__ uIdx, const int64_t* __restrict__ iIdx,
        const float* __restrict__ Gu, const float* __restrict__ Gi, int nu,
        const float* __restrict__ dinv1, float* __restrict__ emb, int nInter) {
    int wv = threadIdx.x >> 5, lane = threadIdx.x & 31;
    int j = blockIdx.x * (blockDim.x >> 5) + wv;
    if (j >= nInter) return;
    int u = (int)uIdx[j];
    int i = (int)iIdx[j];
    float norm = dinv1[u] * dinv1[i] * 0.25f;
    int d = lane * 2;
    float2 a = *(const float2*)(x0_row(Gu, Gi, nu, u) + d);
    float2 b = *(const float2*)(x0_row(Gu, Gi, nu, i) + d);
    float* ei = emb + (size_t)i * 64 + d;
    float* eu = emb + (size_t)u * 64 + d;
    atomicAdd(ei + 0, norm * a.x);   // msg u->i
    atomicAdd(ei + 1, norm * a.y);
    atomicAdd(eu + 0, norm * b.x);   // msg i->u
    atomicAdd(eu + 1, norm * b.y);
}

// ---- routing attention: ui/iu dot products via WMMA segmented reduction ----
// One wave handles 16 interactions. A[j, K] = elementwise products for one
// 4-dim K-block of intent k; B is an indicator (ones in column k only, which
// is invariant to the undocumented K-row striping of the 4x16 B layout), so
// 16 chained WMMAs accumulate D[j,k] = ui[j,k] directly. All selects are
// branch-free so EXEC stays full across the WMMAs.
__global__ void __launch_bounds__(256)
k_inter_wmma(const int64_t* __restrict__ uIdx, const int64_t* __restrict__ iIdx,
             const float* __restrict__ Gu, const float* __restrict__ Gi, int nu,
             const float* __restrict__ emb, float* __restrict__ s2, int nInter) {
    __shared__ float red[8][2][16][16];
#if USE_ASYNC_LDS
    __shared__ int64_t sidx[8][2][16];
#endif
    const int lane = threadIdx.x & 31;
    const int wv   = threadIdx.x >> 5;
    const int jj   = lane & 15;   // interaction slot (matrix row M)
    const int hi   = lane >> 4;   // half-wave: K sub-pair of A layout
    const long long base = ((long long)blockIdx.x * 8 + wv) * 16;
    const long long j = base + jj;

#if USE_ASYNC_LDS
    // Wave-local async staging: each lane DMAs one int64 index into LDS.
    {
        long long src = base + jj;
        if (src > (long long)nInter - 1) src = nInter - 1;   // clamp, stay in-bounds
        const int64_t* g = (lane < 16) ? (uIdx + src) : (iIdx + src);
        int64_t* l = &sidx[wv][lane >> 4][jj];
        __builtin_amdgcn_global_load_async_to_lds_b64(
            (__attribute__((address_space(1))) async_b64_t*)(uintptr_t)g,
            (__attribute__((address_space(3))) async_b64_t*)(uint32_t)(uintptr_t)l,
            0, 0);
        asm volatile("s_wait_asynccnt 0x0" ::: "memory");
    }
    const int u = (int)sidx[wv][0][jj];
    const int i = (int)sidx[wv][1][jj];
#else
    const long long jc = (j < nInter) ? j : (long long)nInter - 1;
    const int u = (int)uIdx[jc];
    const int i = (int)iIdx[jc];
#endif
    const float* xu = x0_row(Gu, Gi, nu, u);
    const float* xi = x0_row(Gu, Gi, nu, i);
    const float* eu = emb + (size_t)u * 64;
    const float* ei = emb + (size_t)i * 64;

    v8f acc_ui = {0.f, 0.f, 0.f, 0.f, 0.f, 0.f, 0.f, 0.f};
    v8f acc_iu = {0.f, 0.f, 0.f, 0.f, 0.f, 0.f, 0.f, 0.f};
#pragma unroll
    for (int k = 0; k < 4; ++k) {
        const float bv = (jj == k) ? 1.0f : 0.0f;   // indicator column k
        v2f B; B[0] = bv; B[1] = bv;
#pragma unroll
        for (int b = 0; b < 4; ++b) {
            // A layout (16x4 f32): lane holds row jj, K = 2*hi + {0,1}
            const int d = k * 16 + b * 4 + hi * 2;
            float2 e_u = *(const float2*)(eu + d);
            float2 e_i = *(const float2*)(ei + d);
            float2 x_u = *(const float2*)(xu + d);
            float2 x_i = *(const float2*)(xi + d);
            v2f Aui, Aiu;
            Aui[0] = e_u.x * fast_tanh(x_i.x);
            Aui[1] = e_u.y * fast_tanh(x_i.y);
            Aiu[0] = e_i.x * fast_tanh(x_u.x);
            Aiu[1] = e_i.y * fast_tanh(x_u.y);
            acc_ui = __builtin_amdgcn_wmma_f32_16x16x4_f32(
                false, Aui, false, B, (short)0, acc_ui, false, false);
            acc_iu = __builtin_amdgcn_wmma_f32_16x16x4_f32(
                false, Aiu, false, B, (short)0, acc_iu, false, false);
        }
    }
    // D layout: VGPR r, lanes 0-15 -> M=r, lanes 16-31 -> M=r+8; N = lane%16
#pragma unroll
    for (int r = 0; r < 8; ++r) {
        red[wv][0][r + 8 * hi][jj] = acc_ui[r];
        red[wv][1][r + 8 * hi][jj] = acc_iu[r];
    }
    __syncthreads();
    if (j < nInter) {
        // lane(jj,hi) finalizes interaction jj, direction hi (0:u->i, 1:i->u)
        float v0 = red[wv][hi][jj][0] + 0.25f;
        float v1 = red[wv][hi][jj][1] + 0.25f;
        float v2 = red[wv][hi][jj][2] + 0.25f;
        float v3 = red[wv][hi][jj][3] + 0.25f;
        float mx = fmaxf(fmaxf(v0, v1), fmaxf(v2, v3));
        float e0 = __expf(v0 - mx), e1 = __expf(v1 - mx);
        float e2 = __expf(v2 - mx), e3 = __expf(v3 - mx);
        float inv = 1.0f / (e0 + e1 + e2 + e3);
        size_t eo = (size_t)(hi ? (j + nInter) : j) * 4;
        s2[eo + 0] = e0 * inv;
        s2[eo + 1] = e1 * inv;
        s2[eo + 2] = e2 * inv;
        s2[eo + 3] = e3 * inv;
    }
}

// ---- per-intent weighted degree for iteration 2 ----
__global__ void k_deg2(const int64_t* __restrict__ uIdx,
                       const int64_t* __restrict__ iIdx,
                       const float* __restrict__ s2,
                       float* __restrict__ deg4, int nInter) {
    int t = blockIdx.x * blockDim.x + threadIdx.x;
    if (t >= nInter * 4) return;
    int j = t >> 2, k = t & 3;
    int u = (int)uIdx[j];
    int i = (int)iIdx[j];
    float sf = s2[(size_t)j * 4 + k];                 // edge u->i (col = i)
    float sb = s2[((size_t)j + nInter) * 4 + k];      // edge i->u (col = u)
    atomicAdd(&deg4[(size_t)i * 4 + k], sf);
    atomicAdd(&deg4[(size_t)u * 4 + k], sb);
}

// ---- propagation, iteration 2 (per-intent norms) ----
__global__ void __launch_bounds__(256)
k_prop2(const int64_t* __restrict__ uIdx, const int64_t* __restrict__ iIdx,
        const float* __restrict__ Gu, const float* __restrict__ Gi, int nu,
        const float* __restrict__ dinv4, const float* __restrict__ s2,
        float* __restrict__ emb, int nInter) {
    int wv = threadIdx.x >> 5, lane = threadIdx.x & 31;
    int j = blockIdx.x * (blockDim.x >> 5) + wv;
    if (j >= nInter) return;
    int u = (int)uIdx[j];
    int i = (int)iIdx[j];
    __builtin_prefetch(s2 + (size_t)j * 4 + 4096, 0, 3);
    int k = lane >> 3;           // intent of this lane's 2 dims
    int d = lane * 2;
    float du = dinv4[(size_t)u * 4 + k];
    float di = dinv4[(size_t)i * 4 + k];
    float n1 = du * di * s2[(size_t)j * 4 + k];             // u->i
    float n2 = du * di * s2[((size_t)j + nInter) * 4 + k];  // i->u
    float2 a = *(const float2*)(x0_row(Gu, Gi, nu, u) + d);
    float2 b = *(const float2*)(x0_row(Gu, Gi, nu, i) + d);
    float* ei = emb + (size_t)i * 64 + d;
    float* eu = emb + (size_t)u * 64 + d;
    atomicAdd(ei + 0, n1 * a.x);
    atomicAdd(ei + 1, n1 * a.y);
    atomicAdd(eu + 0, n2 * b.x);
    atomicAdd(eu + 1, n2 * b.y);
}

// ---- total = ego + emb2 (float4 vectorized) ----
__global__ void k_final(const float* __restrict__ Gu, const float* __restrict__ Gi,
                        int nu, const float* __restrict__ emb,
                        float* __restrict__ out, int nTot4) {
    int t = blockIdx.x * blockDim.x + threadIdx.x;
    if (t >= nTot4) return;
    int n  = t >> 4;     // 16 float4 per 64-dim row
    int d4 = t & 15;
    float4 x = ((const float4*)x0_row(Gu, Gi, nu, n))[d4];
    float4 e = ((const float4*)emb)[t];
    float4 r;
    r.x = x.x + e.x; r.y = x.y + e.y; r.z = x.z + e.z; r.w = x.w + e.w;
    ((float4*)out)[t] = r;
}

extern "C" void kernel_launch(void* const* d_in, const int* in_sizes, int n_in,
                              void* d_out, int out_size, void* d_ws, size_t ws_size,
                              hipStream_t stream) {
    const float*   Gu   = (const float*)d_in[0];
    const float*   Gi   = (const float*)d_in[1];
    const int64_t* edge = (const int64_t*)d_in[2];
    const int nu     = in_sizes[0] / 64;
    const int ni     = in_sizes[1] / 64;
    const int nInter = in_sizes[2] / 4;   // edge_index has 2*(2*nInter) elems
    const int nNodes = nu + ni;
    // edge_index rows: row0 = [u, i], row1 = [i, u]; first half gives (u, i)
    const int64_t* uIdx = edge;
    const int64_t* iIdx = edge + (size_t)2 * nInter;

    char*  ws  = (char*)d_ws;
    size_t off = 0;
    auto alloc = [&](size_t bytes) {
        void* p = ws + off;
        off += (bytes + 255) & ~(size_t)255;
        return p;
    };
    float* emb  = (float*)alloc((size_t)nNodes * 64 * sizeof(float)); // 38.4 MB
    float* s2   = (float*)alloc((size_t)nInter * 8 * sizeof(float));  // 32.0 MB
    float* deg4 = (float*)alloc((size_t)nNodes * 4 * sizeof(float));  //  2.4 MB
    float* deg1 = (float*)alloc((size_t)nNodes * sizeof(float));      //  0.6 MB
    (void)ws_size; (void)n_in; (void)ni;

    const int T = 256;
    // ---- routing iteration 1 ----
    (void)hipMemsetAsync(deg1, 0, (size_t)nNodes * sizeof(float), stream);
    (void)hipMemsetAsync(emb, 0, (size_t)nNodes * 64 * sizeof(float), stream);
    k_degcnt<<<(nInter + T - 1) / T, T, 0, stream>>>(uIdx, iIdx, deg1, nInter);
    k_rsqrt<<<(nNodes + T - 1) / T, T, 0, stream>>>(deg1, nNodes, 0.25f);
    k_prop1<<<(nInter + 7) / 8, T, 0, stream>>>(uIdx, iIdx, Gu, Gi, nu,
                                                deg1, emb, nInter);
    // ---- attention update -> s2 for all 2M directed edges ----
    k_inter_wmma<<<(nInter + 127) / 128, T, 0, stream>>>(uIdx, iIdx, Gu, Gi, nu,
                                                         emb, s2, nInter);
    // ---- routing iteration 2 ----
    (void)hipMemsetAsync(deg4, 0, (size_t)nNodes * 4 * sizeof(float), stream);
    k_deg2<<<(nInter * 4 + T - 1) / T, T, 0, stream>>>(uIdx, iIdx, s2, deg4, nInter);
    k_rsqrt<<<(nNodes * 4 + T - 1) / T, T, 0, stream>>>(deg4, nNodes * 4, 1.0f);
    (void)hipMemsetAsync(emb, 0, (size_t)nNodes * 64 * sizeof(float), stream);
    k_prop2<<<(nInter + 7) / 8, T, 0, stream>>>(uIdx, iIdx, Gu, Gi, nu,
                                                deg4, s2, emb, nInter);
    // ---- output ----
    k_final<<<(out_size / 4 + T - 1) / T, T, 0, stream>>>(Gu, Gi, nu, emb,
                                                          (float*)d_out,
                                                          out_size / 4);
}